// GraphConv_81913616269702
// MI455X (gfx1250) — compile-verified
//
#include <hip/hip_runtime.h>

// GraphConv: out = relu( segment_sum(val * (X@W)[src] -> dst) + b )
// N_NODES=50000, N_EDGES=800000, D_IN=D_OUT=256

#define D 256            // D_IN == D_OUT
#define TILE 16
#define CTILE 64         // output columns per wave (4 WMMA tiles, shared A)
#define WAVE 32

typedef __attribute__((ext_vector_type(2))) float v2f;
typedef __attribute__((ext_vector_type(8))) float v8f;

// ---------------------------------------------------------------------------
// Kernel 1: h = X @ W via V_WMMA_F32_16X16X4_F32.
// One wave computes a 16x64 output stripe: the A fragment (16x4 fp32, one
// global_load_b64 per lane per k-step) is reused across 4 column tiles.
// ---------------------------------------------------------------------------
__global__ void gemm_xw_wmma(const float* __restrict__ x,
                             const float* __restrict__ W,
                             float* __restrict__ h,
                             int total_waves) {
  const int wave = (blockIdx.x * blockDim.x + threadIdx.x) >> 5;
  if (wave >= total_waves) return;          // wave-uniform: EXEC stays all-ones
  const int lane   = threadIdx.x & 31;
  const int groupsN = D / CTILE;            // 4
  const int tm = wave / groupsN;
  const int tg = wave - tm * groupsN;
  const int row0 = tm * TILE;
  const int col0 = tg * CTILE;

  const int m     = lane & 15;              // A: row in tile; B: col in tile
  const int khalf = (lane >> 4) << 1;       // lanes 0-15 -> K+0/1, 16-31 -> K+2/3

  const float* arow  = x + (size_t)(row0 + m) * D + khalf;  // contiguous pair
  const float* bbase = W + col0 + m;

  v8f c0 = {}, c1 = {}, c2 = {}, c3 = {};
#pragma unroll 4
  for (int kk = 0; kk < D; kk += 4) {
    v2f a = *(const v2f*)(arow + kk);                       // global_load_b64
    const float* brow0 = bbase + (size_t)(kk + khalf) * D;
    const float* brow1 = brow0 + D;
    v2f b0 = {brow0[ 0], brow1[ 0]};
    v2f b1 = {brow0[16], brow1[16]};
    v2f b2 = {brow0[32], brow1[32]};
    v2f b3 = {brow0[48], brow1[48]};
    // 8 args: (neg_a, A, neg_b, B, c_mod, C, reuse_a, reuse_b)
    c0 = __builtin_amdgcn_wmma_f32_16x16x4_f32(false, a, false, b0, (short)0, c0, false, false);
    c1 = __builtin_amdgcn_wmma_f32_16x16x4_f32(false, a, false, b1, (short)0, c1, false, false);
    c2 = __builtin_amdgcn_wmma_f32_16x16x4_f32(false, a, false, b2, (short)0, c2, false, false);
    c3 = __builtin_amdgcn_wmma_f32_16x16x4_f32(false, a, false, b3, (short)0, c3, false, false);
  }

  // C/D layout: VGPR r -> (M=r, N=lane) lanes 0-15, (M=r+8, N=lane-16) lanes 16-31
  const int mbase = row0 + ((lane >> 4) << 3);
  float* hp = h + (size_t)mbase * D + col0 + m;
#pragma unroll
  for (int r = 0; r < 8; ++r) {
    hp[(size_t)r * D +  0] = c0[r];
    hp[(size_t)r * D + 16] = c1[r];
    hp[(size_t)r * D + 32] = c2[r];
    hp[(size_t)r * D + 48] = c3[r];
  }
}

// ---------------------------------------------------------------------------
// Kernel 2: zero the accumulator (d_out), vectorized b128 stores.
// ---------------------------------------------------------------------------
__global__ void zero_f4(float4* __restrict__ p, int n4) {
  int t = blockIdx.x * blockDim.x + threadIdx.x;
  if (t < n4) p[t] = make_float4(0.f, 0.f, 0.f, 0.f);
}

// ---------------------------------------------------------------------------
// Kernel 3: COO scatter. One wave per edge; each lane owns 8 channels.
// Gathers of h hit the 192MB L2 (h is 51.2MB); non-returning f32 atomics.
// ---------------------------------------------------------------------------
__global__ void scatter_edges(const int* __restrict__ esrc,
                              const int* __restrict__ edst,
                              const float* __restrict__ eval,
                              const float* __restrict__ h,
                              float* __restrict__ z,
                              int n_edges) {
  int t = blockIdx.x * blockDim.x + threadIdx.x;
  int e = t >> 5;
  if (e >= n_edges) return;
  int lane = t & 31;

  int   s = esrc[e];
  int   d = edst[e];
  float v = eval[e];

  const float4* hs = (const float4*)(h + (size_t)s * D) + lane * 2;
  float*        zd = z + (size_t)d * D + lane * 8;

  float4 h0 = hs[0];                        // global_load_b128 x2 (wave reads 1KB contiguous)
  float4 h1 = hs[1];

  atomicAdd(zd + 0, v * h0.x);
  atomicAdd(zd + 1, v * h0.y);
  atomicAdd(zd + 2, v * h0.z);
  atomicAdd(zd + 3, v * h0.w);
  atomicAdd(zd + 4, v * h1.x);
  atomicAdd(zd + 5, v * h1.y);
  atomicAdd(zd + 6, v * h1.z);
  atomicAdd(zd + 7, v * h1.w);
}

// ---------------------------------------------------------------------------
// Kernel 4: out = relu(out + b), in place, vectorized.
// ---------------------------------------------------------------------------
__global__ void bias_relu(float4* __restrict__ z, const float4* __restrict__ b4,
                          int n4) {
  int t = blockIdx.x * blockDim.x + threadIdx.x;
  if (t >= n4) return;
  float4 zv = z[t];
  float4 bv = b4[t & (D / 4 - 1)];          // row length 256 -> 64 float4 per row
  zv.x = fmaxf(zv.x + bv.x, 0.f);
  zv.y = fmaxf(zv.y + bv.y, 0.f);
  zv.z = fmaxf(zv.z + bv.z, 0.f);
  zv.w = fmaxf(zv.w + bv.w, 0.f);
  z[t] = zv;
}

// ---------------------------------------------------------------------------
extern "C" void kernel_launch(void* const* d_in, const int* in_sizes, int n_in,
                              void* d_out, int out_size, void* d_ws, size_t ws_size,
                              hipStream_t stream) {
  const int*   edge_src = (const int*)d_in[0];
  const int*   edge_dst = (const int*)d_in[1];
  const float* edge_val = (const float*)d_in[2];
  const float* x        = (const float*)d_in[3];
  const float* W        = (const float*)d_in[4];
  const float* b        = (const float*)d_in[5];
  float*       out      = (float*)d_out;

  const int n_edges = in_sizes[0];
  const int n_nodes = in_sizes[3] / D;

  float* h = (float*)d_ws;                  // [n_nodes, D] fp32, 51.2 MB

  // Zero accumulator (independent of GEMM; stream-ordered before scatter).
  {
    int n4 = (n_nodes * D) / 4;
    int blocks = (n4 + 255) / 256;
    zero_f4<<<blocks, 256, 0, stream>>>((float4*)out, n4);
  }

  // h = X @ W  (WMMA fp32, 16x64 per wave)
  {
    int total_waves = (n_nodes / TILE) * (D / CTILE);  // 3125 * 4 = 12500
    int blocks = (total_waves * WAVE + 255) / 256;     // 8 waves per block
    gemm_xw_wmma<<<blocks, 256, 0, stream>>>(x, W, h, total_waves);
  }

  // z[dst] += val * h[src]
  {
    long long threads = (long long)n_edges * WAVE;
    int blocks = (int)((threads + 255) / 256);
    scatter_edges<<<blocks, 256, 0, stream>>>(edge_src, edge_dst, edge_val,
                                              h, out, n_edges);
  }

  // out = relu(out + b)
  {
    int n4 = (n_nodes * D) / 4;
    int blocks = (n4 + 255) / 256;
    bias_relu<<<blocks, 256, 0, stream>>>((float4*)out, (const float4*)b, n4);
  }
}